// MockStreamGenerator_8083128451470
// MI455X (gfx1250) — compile-verified
//
#include <hip/hip_runtime.h>
#include <stdint.h>

// Hernquist stream generator for MI455X (gfx1250, wave32).
//   GM = 1, C = 1, N_STEPS = 128, N_SUB = 8.
// Particle integration: one particle per lane, all state in VGPRs.
// Progenitor orbit: serial dependency chain (the runtime bottleneck);
//   dt table staged in LDS, chain shortened with 1-instruction
//   v_sqrt_f32 / v_rcp_f32 (<=1 ulp; error enters scaled by dt -> ~1e-6 final).
// CDNA5 data path: global_load_async_to_lds_b128 + s_wait_asynccnt (ASYNCcnt).

#define N_STEPS 128
#define N_SUB   8

struct W6 { float qx, qy, qz, px, py, pz; };

__device__ __forceinline__ W6 deriv(const W6& w) {
    float r2  = fmaf(w.qx, w.qx, fmaf(w.qy, w.qy, w.qz * w.qz));
    float r   = __builtin_amdgcn_sqrtf(r2);        // v_sqrt_f32, <=1 ulp
    float rc  = r + 1.0f;                          // r + C, C = 1
    float den = fmaf(r * rc, rc, 1e-12f);          // r*(r+C)^2 + 1e-12
    float s   = -__builtin_amdgcn_rcpf(den);       // v_rcp_f32, <=1 ulp; -GM/den
    W6 k;
    k.qx = w.px;     k.qy = w.py;     k.qz = w.pz;
    k.px = w.qx * s; k.py = w.qy * s; k.pz = w.qz * s;
    return k;
}

__device__ __forceinline__ W6 axpy(const W6& w, float a, const W6& k) {
    W6 r;
    r.qx = fmaf(a, k.qx, w.qx); r.qy = fmaf(a, k.qy, w.qy); r.qz = fmaf(a, k.qz, w.qz);
    r.px = fmaf(a, k.px, w.px); r.py = fmaf(a, k.py, w.py); r.pz = fmaf(a, k.pz, w.pz);
    return r;
}

__device__ __forceinline__ W6 rk4(const W6& w, float dt) {
    W6 k1 = deriv(w);
    W6 k2 = deriv(axpy(w, 0.5f * dt, k1));
    W6 k3 = deriv(axpy(w, 0.5f * dt, k2));
    W6 k4 = deriv(axpy(w, dt, k3));
    const float c = dt * (1.0f / 6.0f);
    W6 r;
    r.qx = fmaf(c, fmaf(2.0f, k2.qx + k3.qx, k1.qx + k4.qx), w.qx);
    r.qy = fmaf(c, fmaf(2.0f, k2.qy + k3.qy, k1.qy + k4.qy), w.qy);
    r.qz = fmaf(c, fmaf(2.0f, k2.qz + k3.qz, k1.qz + k4.qz), w.qz);
    r.px = fmaf(c, fmaf(2.0f, k2.px + k3.px, k1.px + k4.px), w.px);
    r.py = fmaf(c, fmaf(2.0f, k2.py + k3.py, k1.py + k4.py), w.py);
    r.pz = fmaf(c, fmaf(2.0f, k2.pz + k3.pz, k1.pz + k4.pz), w.pz);
    return r;
}

// CDNA5 async global->LDS copy (tracked by ASYNCcnt; see CDNA5 ISA).
__device__ __forceinline__ void async_g2l_b128(uint32_t lds_byte_off, const void* gaddr) {
    asm volatile("global_load_async_to_lds_b128 %0, %1, off"
                 :
                 : "v"(lds_byte_off), "v"(gaddr)
                 : "memory");
}
__device__ __forceinline__ void wait_asynccnt0() {
    asm volatile("s_wait_asynccnt 0" ::: "memory");
}

__global__ __launch_bounds__(256) void stream_rk4_kernel(
    const float* __restrict__ ts,        // [n]
    const float* __restrict__ prog_w0,   // [6]
    const float* __restrict__ w0_lead,   // [n,6]
    const float* __restrict__ w0_trail,  // [n,6]
    const float* __restrict__ rel_lead,  // [n]
    const float* __restrict__ rel_trail, // [n]
    float* __restrict__ out,             // 22n floats: q(6n) p(6n) t(2n) rel(2n) prog(6n)
    int n)
{
    __shared__ float smem[8192];  // 32 KB: dt table (progenitor block) or w0 tile (particle blocks)
    const int   tid    = threadIdx.x;
    const float t_last = ts[n - 1];
    const float t_f    = t_last + 0.001f;

    if ((int)blockIdx.x == (int)gridDim.x - 1) {
        // ---------------- Progenitor orbit (serial critical path) ----------------
        // Cooperatively stage all segment dts into LDS so the serial loop never
        // waits on global memory.
        for (int i = tid; i < n - 1; i += 256)
            smem[i] = (ts[i + 1] - ts[i]) * (1.0f / N_SUB);
        __syncthreads();

        if (tid == 0) {
            float* po = out + (size_t)16 * n;  // prog_orbit base
            W6 w;
            w.qx = prog_w0[0]; w.qy = prog_w0[1]; w.qz = prog_w0[2];
            w.px = prog_w0[3]; w.py = prog_w0[4]; w.pz = prog_w0[5];
            po[0] = w.qx; po[1] = w.qy; po[2] = w.qz;
            po[3] = w.px; po[4] = w.py; po[5] = w.pz;
            #pragma unroll 1
            for (int seg = 0; seg < n - 1; ++seg) {
                const float dt = smem[seg];
                #pragma unroll
                for (int s = 0; s < N_SUB; ++s) w = rk4(w, dt);
                float* row = po + (size_t)(seg + 1) * 6;
                row[0] = w.qx; row[1] = w.qy; row[2] = w.qz;
                row[3] = w.px; row[4] = w.py; row[5] = w.pz;
            }
        }
        return;
    }

    // ---------------- Stream particles: one particle per lane ----------------
    // Block b covers 256 consecutive particles of one half (n == 8192 is a
    // multiple of 256, so halves split exactly on block boundaries).
    const int  b       = (int)blockIdx.x;
    const int  nb_half = n >> 8;              // blocks per half
    const bool is_lead = (b < nb_half);
    const int  hb      = is_lead ? b : (b - nb_half);
    const float* src   = (is_lead ? w0_lead : w0_trail) + (size_t)hb * 1536;

    // Stage this block's 1536 floats (6 KB) of initial conditions via the
    // CDNA5 async engine: 384 b128 transfers, per-lane LDS/global addresses.
    {
        const uint32_t lds_base = (uint32_t)(uintptr_t)&smem[0];
        const float4*  s4       = (const float4*)src;
        async_g2l_b128(lds_base + 16u * (uint32_t)tid, s4 + tid);
        if (tid < 128)
            async_g2l_b128(lds_base + 16u * (uint32_t)(256 + tid), s4 + 256 + tid);
        wait_asynccnt0();
        __syncthreads();
    }

    const int   i  = hb * 256 + tid;                 // particle index within half
    const float t0 = ts[i];
    const float dt = (t_f - t0) * (1.0f / N_STEPS);  // /128 exact

    const float* ws = &smem[tid * 6];  // stride 6 banks: conflict-free for 32 lanes
    W6 w;
    w.qx = ws[0]; w.qy = ws[1]; w.qz = ws[2];
    w.px = ws[3]; w.py = ws[4]; w.pz = ws[5];

    #pragma unroll 1
    for (int s = 0; s < N_STEPS; ++s) w = rk4(w, dt);

    // interleave(trail, lead): trail -> row 2i, lead -> row 2i+1
    const int row = is_lead ? (2 * i + 1) : (2 * i);
    float* outq = out;
    float* outp = out + (size_t)6 * n;
    float* outt = out + (size_t)12 * n;
    float* outr = out + (size_t)14 * n;

    outq[row * 3 + 0] = w.qx; outq[row * 3 + 1] = w.qy; outq[row * 3 + 2] = w.qz;
    outp[row * 3 + 0] = w.px; outp[row * 3 + 1] = w.py; outp[row * 3 + 2] = w.pz;
    outt[row] = t_last;
    // release_time = interleave(release_lead, release_trail): lead -> 2i, trail -> 2i+1
    if (is_lead) outr[2 * i]     = rel_lead[i];
    else         outr[2 * i + 1] = rel_trail[i];
}

extern "C" void kernel_launch(void* const* d_in, const int* in_sizes, int n_in,
                              void* d_out, int out_size, void* d_ws, size_t ws_size,
                              hipStream_t stream) {
    const float* ts        = (const float*)d_in[0];
    const float* prog_w0   = (const float*)d_in[1];
    const float* w0_lead   = (const float*)d_in[2];
    const float* w0_trail  = (const float*)d_in[3];
    const float* rel_lead  = (const float*)d_in[4];
    const float* rel_trail = (const float*)d_in[5];
    float*       out       = (float*)d_out;
    const int    n         = in_sizes[0];  // 8192

    // 2n/256 particle blocks + 1 progenitor block, fused so the serial
    // progenitor chain overlaps the particle work instead of serializing.
    const int nblocks = (2 * n) / 256 + 1;
    hipLaunchKernelGGL(stream_rk4_kernel, dim3(nblocks), dim3(256), 0, stream,
                       ts, prog_w0, w0_lead, w0_trail, rel_lead, rel_trail, out, n);
}